// HierarchicalWaveFieldKV_30356828848214
// MI455X (gfx1250) — compile-verified
//
#include <hip/hip_runtime.h>

#define TWO_PI_F 6.28318530717958647692f
#define INV_MSL  (1.0f/8192.0f)

#define B_  16
#define S_  4096
#define P_  4096
#define D_  128

typedef float v2f __attribute__((ext_vector_type(2)));
typedef float v8f __attribute__((ext_vector_type(8)));
typedef int v4i_b128 __attribute__((vector_size(16)));  // matches builtin param

#define AS_GLOBAL __attribute__((address_space(1)))
#define AS_LOCAL  __attribute__((address_space(3)))

// ---- gfx1250 async LDS<->global data movement (ASYNCcnt), probe-guarded ----
#if defined(__has_builtin)
#  if __has_builtin(__builtin_amdgcn_global_load_async_to_lds_b128) && \
      __has_builtin(__builtin_amdgcn_s_wait_asynccnt)
#    define HWF_ASYNC_LOAD 1
#  endif
#  if __has_builtin(__builtin_amdgcn_global_store_async_from_lds_b128) && \
      __has_builtin(__builtin_amdgcn_s_wait_asynccnt)
#    define HWF_ASYNC_STORE 1
#  endif
#endif
#ifndef HWF_ASYNC_LOAD
#  define HWF_ASYNC_LOAD 0
#endif
#ifndef HWF_ASYNC_STORE
#  define HWF_ASYNC_STORE 0
#endif

// D = A(16x4) * B(4x16) + C, f32 WMMA (CDNA5, wave32)
__device__ __forceinline__ v8f wmma_f32x4(v2f a, v2f b, v8f c) {
  return __builtin_amdgcn_wmma_f32_16x16x4_f32(false, a, false, b, (short)0, c,
                                               false, false);
}

// ---- per-level compile-time metadata (LEVELS in reference) ----
__device__ __forceinline__ int level_n(int l) {
  return l == 0 ? 16 : l == 1 ? 32 : l == 2 ? 64 : l == 3 ? 128 : 256;
}
__device__ __forceinline__ float level_alpha(int l) {
  return l == 0 ? 0.001f : l == 1 ? 0.005f : l == 2 ? 0.02f : l == 3 ? 0.05f : 0.1f;
}
__device__ __forceinline__ int enc_off(int l) {   // cumsum of 3n*128
  return l == 0 ? 0 : l == 1 ? 6144 : l == 2 ? 18432 : l == 3 ? 43008 : 92160;
}
__device__ __forceinline__ int dec_off(int l) {   // cumsum of n*128
  return l == 0 ? 0 : l == 1 ? 2048 : l == 2 ? 6144 : l == 3 ? 14336 : 30720;
}
__device__ __forceinline__ int ema_off(int l) {   // cumsum of B*3n
  return l == 0 ? 0 : l == 1 ? 768 : l == 2 ? 2304 : l == 3 ? 5376 : 11520;
}

// Stage a 16x128 f32 tile (row stride 132 in LDS, bank-conflict free).
__device__ __forceinline__ void stage_tile(float* dstLds, const float* srcGlb,
                                           int lane) {
#if HWF_ASYNC_LOAD
  #pragma unroll
  for (int j = 0; j < 16; ++j) {
    __builtin_amdgcn_global_load_async_to_lds_b128(
        (AS_GLOBAL v4i_b128*)(srcGlb + j * D_ + lane * 4),
        (AS_LOCAL v4i_b128*)(dstLds + j * 132 + lane * 4), 0, 0);
  }
  asm volatile("" ::: "memory");
#else
  float4 tmp[16];
  #pragma unroll
  for (int j = 0; j < 16; ++j)
    tmp[j] = *(const float4*)(srcGlb + j * D_ + lane * 4);
  #pragma unroll
  for (int j = 0; j < 16; ++j)
    *(float4*)(dstLds + j * 132 + lane * 4) = tmp[j];
#endif
}

__device__ __forceinline__ void wait_async_le16() {
#if HWF_ASYNC_LOAD
  asm volatile("" ::: "memory");
  __builtin_amdgcn_s_wait_asynccnt(16);
  asm volatile("" ::: "memory");
#endif
}
__device__ __forceinline__ void wait_async_0() {
#if HWF_ASYNC_LOAD || HWF_ASYNC_STORE
  asm volatile("" ::: "memory");
  __builtin_amdgcn_s_wait_asynccnt(0);
  asm volatile("" ::: "memory");
#endif
}

// ---------------------------------------------------------------------------
// Kernel 0: generic transpose  dst[c*R + r] = src[r*C + c]
// ---------------------------------------------------------------------------
__global__ void hwf_transpose_kernel(const float* __restrict__ src,
                                     float* __restrict__ dst, int R, int C) {
  int i = blockIdx.x * blockDim.x + threadIdx.x;
  if (i < R * C) {
    int r = i / C, c = i - r * C;
    dst[c * R + r] = src[i];
  }
}

// ---------------------------------------------------------------------------
// Kernel 1: fused encode-GEMM + EMA reduction.
// grid = B * 93 (93 = total 16-wide w-tiles over all levels), block = 256.
// Each WG: one (b, level, w-tile). Each wave: 32 s-tiles, double-buffered
// through its private LDS region via async load-to-LDS.
// ---------------------------------------------------------------------------
__global__ void __launch_bounds__(256)
hwf_ema_encode_kernel(const float* __restrict__ kv,
                      const int* __restrict__ positions,
                      const float* __restrict__ encT,  // [D, 3n] per level
                      float* __restrict__ ema,         // [B, 3n] per level
                      const float* __restrict__ bf0, const float* __restrict__ bf1,
                      const float* __restrict__ bf2, const float* __restrict__ bf3,
                      const float* __restrict__ bf4) {
  const int blk = blockIdx.x;
  const int b = blk / 93;
  const int wg = blk - b * 93;
  const int l = (wg < 3) ? 0 : (wg < 9) ? 1 : (wg < 21) ? 2 : (wg < 45) ? 3 : 4;
  const int wt = wg - ((l == 0) ? 0 : l == 1 ? 3 : l == 2 ? 9 : l == 3 ? 21 : 45);
  const int n = level_n(l);
  const int W = 3 * n;
  const int w0 = wt * 16;
  const int type = w0 / n;  // 0=freq, 1=amp, 2=phase (tile never straddles)

  const float alpha = level_alpha(l);
  const float l2a = __log2f(1.0f - alpha);
  const float* encT_l = encT + enc_off(l);
  const float* basef = (l == 0) ? bf0 : (l == 1) ? bf1 : (l == 2) ? bf2
                                      : (l == 3) ? bf3 : bf4;

  const int tid = threadIdx.x;
  const int wv = tid >> 5;
  const int lane = tid & 31;
  const int lrow = lane & 15;
  const int koff = (lane >> 4) << 1;  // K offset: 0 (lanes 0-15) / 2 (16-31)
  const int moff = (lane >> 4) << 3;  // C-tile row offset for high lanes

  // B tile (W_encT 128 x 16) cached in registers: reused across 32 s-tiles.
  v2f Breg[32];
  {
    const int col = w0 + lrow;
    #pragma unroll
    for (int j = 0; j < 32; ++j) {
      const int kk = 4 * j + koff;
      Breg[j].x = encT_l[kk * W + col];
      Breg[j].y = encT_l[(kk + 1) * W + col];
    }
  }
  const float bft = (type == 0) ? basef[w0 + lrow] : 0.0f;

  // per-wave double-buffered LDS tile, row stride 132 floats
  __shared__ __align__(16) float ldsA[8][2 * 2112];
  __shared__ float red[8][16];
  float* bufA = ldsA[wv];

  const float* srcBase = kv + ((size_t)b * S_ + (size_t)wv * 32 * 16) * D_;
  stage_tile(bufA, srcBase, lane);  // prologue: tile 0 into buffer 0

  float acc = 0.0f;

  #pragma unroll 1
  for (int t = 0; t < 32; ++t) {
    const int s0 = (wv * 32 + t) * 16;
    float* cur = bufA + (t & 1) * 2112;
    float* nxt = bufA + ((t + 1) & 1) * 2112;
    const float* srcNext = srcBase + (size_t)(t + 1) * 16 * D_;

    if (t < 31) {
      stage_tile(nxt, srcNext, lane);
      if (t < 30)  // warm L2 for the tile after next (global_prefetch_b8)
        __builtin_prefetch(srcNext + 16 * D_ + lane * 16, 0, 1);
      wait_async_le16();   // tile t's 16 async loads complete; t+1 in flight
    } else {
      wait_async_0();
    }

    // 16x16 delta tile via 32 chained f32 WMMAs over K=D=128
    v8f c = {0.f, 0.f, 0.f, 0.f, 0.f, 0.f, 0.f, 0.f};
    #pragma unroll
    for (int k4 = 0; k4 < 32; ++k4) {
      const float* ap = cur + lrow * 132 + 4 * k4 + koff;
      v2f a;
      a.x = ap[0];
      a.y = ap[1];
      c = wmma_f32x4(a, Breg[k4], c);
    }

    // fused EMA epilogue: weight = alpha*(1-alpha)^(S-1-s)
    #pragma unroll
    for (int m = 0; m < 8; ++m) {
      const int srow = s0 + m + moff;
      const float wgt = alpha * exp2f(l2a * (float)(S_ - 1 - srow));
      float val = c[m];
      if (type == 0) {
        const float pos = (float)positions[srow];
        val *= sinf(TWO_PI_F * bft * (pos * INV_MSL));
      } else if (type == 1) {
        val = fabsf(val);
      }
      acc += wgt * val;
    }
  }

  // lanes L and L+16 hold the same w column (different s rows): combine
  acc += __shfl_xor(acc, 16, 32);
  if (lane < 16) red[wv][lane] = acc;
  __syncthreads();
  if (tid < 16) {
    float s = 0.0f;
    #pragma unroll
    for (int r = 0; r < 8; ++r) s += red[r][tid];
    ema[ema_off(l) + b * W + w0 + tid] = s;
  }
}

// ---------------------------------------------------------------------------
// Kernel 2: wave-field reconstruction, all 5 levels fused in registers.
// grid = B * 32, block = 256 (8 waves). Wave = 16-row p-tile x full D=128.
// Output transposed through LDS -> coalesced b128 stores (async when avail).
// ---------------------------------------------------------------------------
__global__ void __launch_bounds__(256)
hwf_recon_kernel(const int* __restrict__ rpos,
                 const float* __restrict__ decT,  // [n, 128] per level
                 const float* __restrict__ ema,   // [B, 3n] per level
                 const float* __restrict__ bf0, const float* __restrict__ bf1,
                 const float* __restrict__ bf2, const float* __restrict__ bf3,
                 const float* __restrict__ bf4,
                 float* __restrict__ out) {
  const int b = blockIdx.x >> 5;
  const int chunk = blockIdx.x & 31;
  const int tid = threadIdx.x;
  const int wv = tid >> 5;
  const int lane = tid & 31;
  const int lrow = lane & 15;
  const int koff = (lane >> 4) << 1;
  const int moff = (lane >> 4) << 3;
  const int p0 = chunk * 128 + wv * 16;

  const float rp = (float)rpos[p0 + lrow];

  v8f acc[8];
  #pragma unroll
  for (int j = 0; j < 8; ++j) acc[j] = (v8f){0.f, 0.f, 0.f, 0.f, 0.f, 0.f, 0.f, 0.f};

  __shared__ float ampS[256], omgS[256], phS[256];
  __shared__ __align__(16) float ldsO[8][2112];

  #pragma unroll 1
  for (int l = 0; l < 5; ++l) {
    const int n = level_n(l);
    const float* ema_l = ema + ema_off(l) + b * 3 * n;
    const float* basef = (l == 0) ? bf0 : (l == 1) ? bf1 : (l == 2) ? bf2
                                        : (l == 3) ? bf3 : bf4;
    const float* decT_l = decT + dec_off(l);

    for (int i = tid; i < n; i += 256) {
      const float fr = ema_l[i];
      ampS[i] = ema_l[n + i];
      phS[i]  = ema_l[2 * n + i];
      omgS[i] = TWO_PI_F * (fr + basef[i]) * INV_MSL;
    }
    __syncthreads();

    #pragma unroll 1
    for (int k0 = 0; k0 < n; k0 += 4) {
      const int kk = k0 + koff;
      v2f a;
      a.x = ampS[kk]     * sinf(omgS[kk]     * rp + phS[kk]);
      a.y = ampS[kk + 1] * sinf(omgS[kk + 1] * rp + phS[kk + 1]);
      const float* bp = decT_l + kk * D_ + lrow;
      #pragma unroll
      for (int j = 0; j < 8; ++j) {
        v2f bb;
        bb.x = bp[j * 16];
        bb.y = bp[D_ + j * 16];
        acc[j] = wmma_f32x4(a, bb, acc[j]);
      }
    }
    __syncthreads();
  }

  // transpose accumulators through per-wave LDS for coalesced row stores
  float* myO = ldsO[wv];
  #pragma unroll
  for (int j = 0; j < 8; ++j) {
    #pragma unroll
    for (int m = 0; m < 8; ++m) {
      myO[(m + moff) * 132 + j * 16 + lrow] = acc[j][m];
    }
  }

  float* dst = out + (size_t)b * P_ * D_ + (size_t)p0 * D_;
#if HWF_ASYNC_STORE
  // DScnt and ASYNCcnt are independent: make LDS writes land first
  asm volatile("s_wait_dscnt 0x0" ::: "memory");
  #pragma unroll
  for (int j = 0; j < 16; ++j) {
    __builtin_amdgcn_global_store_async_from_lds_b128(
        (AS_GLOBAL v4i_b128*)(dst + j * D_ + lane * 4),
        (AS_LOCAL v4i_b128*)(myO + j * 132 + lane * 4), 0, 0);
  }
  wait_async_0();
#else
  #pragma unroll
  for (int j = 0; j < 16; ++j) {
    float4 vv = *(const float4*)(myO + j * 132 + lane * 4);
    *(float4*)(dst + j * D_ + lane * 4) = vv;
  }
#endif
}

// ---------------------------------------------------------------------------
extern "C" void kernel_launch(void* const* d_in, const int* in_sizes, int n_in,
                              void* d_out, int out_size, void* d_ws, size_t ws_size,
                              hipStream_t stream) {
  const float* kv = (const float*)d_in[0];
  const int* positions = (const int*)d_in[1];
  const int* rpos = (const int*)d_in[2];

  const float* W_enc[5];
  const float* W_dec[5];
  const float* bfs[5];
  for (int i = 0; i < 5; ++i) {
    W_enc[i] = (const float*)d_in[3 + 3 * i];
    W_dec[i] = (const float*)d_in[4 + 3 * i];
    bfs[i]   = (const float*)d_in[5 + 3 * i];
  }

  static const int NS[5] = {16, 32, 64, 128, 256};
  static const int ENC_OFF[5] = {0, 6144, 18432, 43008, 92160};   // 3n*128 cumsum
  static const int DEC_OFF[5] = {0, 2048, 6144, 14336, 30720};    // n*128 cumsum
  const int ENC_TOT = 190464, DEC_TOT = 63488;

  float* ws = (float*)d_ws;
  float* encT = ws;                       // 190464 floats
  float* decT = ws + ENC_TOT;             // 63488 floats
  float* ema  = ws + ENC_TOT + DEC_TOT;   // 23808 floats
  (void)ws_size; (void)n_in; (void)in_sizes; (void)out_size;

  // transpose weights for coalesced WMMA B-fragment loads
  for (int i = 0; i < 5; ++i) {
    int ne = 3 * NS[i] * D_;
    hwf_transpose_kernel<<<(ne + 255) / 256, 256, 0, stream>>>(
        W_enc[i], encT + ENC_OFF[i], 3 * NS[i], D_);
    int nd = D_ * NS[i];
    hwf_transpose_kernel<<<(nd + 255) / 256, 256, 0, stream>>>(
        W_dec[i], decT + DEC_OFF[i], D_, NS[i]);
  }

  // fused encode GEMM + EMA  (16 b * 93 w-tiles)
  hwf_ema_encode_kernel<<<16 * 93, 256, 0, stream>>>(
      kv, positions, encT, ema, bfs[0], bfs[1], bfs[2], bfs[3], bfs[4]);

  // reconstruction GEMM, all levels fused
  hwf_recon_kernel<<<16 * 32, 256, 0, stream>>>(
      rpos, decT, ema, bfs[0], bfs[1], bfs[2], bfs[3], bfs[4], (float*)d_out);
}